// PointNetSetAbstraction_17265768530253
// MI455X (gfx1250) — compile-verified
//
#include <hip/hip_runtime.h>

typedef float v2f __attribute__((ext_vector_type(2)));
typedef float v8f __attribute__((ext_vector_type(8)));

#define NBATCH   16
#define NPTS     4096
#define NPOINTS  1024
#define NSAMP    32
#define MCOLS    (NBATCH * NPOINTS * NSAMP)   // 524288
#define RAD2     0.04f
#define EPSV     1e-5f

// ---- workspace layout (in floats) ----
static constexpr size_t OFF_X0   = 0;                                  // [8][M]
static constexpr size_t OFF_Y0   = OFF_X0 + (size_t)8  * MCOLS;        // [64][M]
static constexpr size_t OFF_Y1   = OFF_Y0 + (size_t)64 * MCOLS;        // [64][M]
static constexpr size_t OFF_W0P  = OFF_Y1 + (size_t)64 * MCOLS;        // [64][8]
static constexpr size_t OFF_SUM0 = OFF_W0P + 512;                      // 512 floats of stats
static constexpr size_t OFF_SQ0  = OFF_SUM0 + 64;
static constexpr size_t OFF_SUM1 = OFF_SQ0 + 64;
static constexpr size_t OFF_SQ1  = OFF_SUM1 + 64;
static constexpr size_t OFF_SUM2 = OFF_SQ1 + 64;
static constexpr size_t OFF_SQ2  = OFF_SUM2 + 128;
static constexpr size_t OFF_NS0  = OFF_SQ2 + 128;
static constexpr size_t OFF_NH0  = OFF_NS0 + 64;
static constexpr size_t OFF_NS1  = OFF_NH0 + 64;
static constexpr size_t OFF_NH1  = OFF_NS1 + 64;
static constexpr size_t OFF_NS2  = OFF_NH1 + 64;
static constexpr size_t OFF_NH2  = OFF_NS2 + 128;

// =====================================================================
// 1) Farthest point sampling: one block per batch, 256 threads,
//    16 points per thread held in registers. Emits new_xyz directly.
// =====================================================================
__global__ __launch_bounds__(256)
void fps_kernel(const float* __restrict__ xyz, float* __restrict__ new_xyz)
{
    const int b    = blockIdx.x;
    const int tid  = threadIdx.x;      // 256
    const int lane = tid & 31;
    const int wv   = tid >> 5;
    const float* xb = xyz + (size_t)b * NPTS * 3;

    float px[16], py[16], pz[16], pd[16];
#pragma unroll
    for (int j = 0; j < 16; ++j) {
        const int p = tid + j * 256;
        px[j] = xb[p * 3 + 0];
        py[j] = xb[p * 3 + 1];
        pz[j] = xb[p * 3 + 2];
        pd[j] = 1e10f;
    }

    __shared__ float rv[8];
    __shared__ int   ri[8];
    __shared__ float scx, scy, scz;
    __shared__ int   sfar;
    if (tid == 0) sfar = 0;
    __syncthreads();

    for (int i = 0; i < NPOINTS; ++i) {
        if (tid == 0) {
            const int f = sfar;
            const float cx = xb[f * 3 + 0];
            const float cy = xb[f * 3 + 1];
            const float cz = xb[f * 3 + 2];
            scx = cx; scy = cy; scz = cz;
            float* o = new_xyz + ((size_t)b * NPOINTS + i) * 3;
            o[0] = cx; o[1] = cy; o[2] = cz;
        }
        __syncthreads();
        const float cx = scx, cy = scy, cz = scz;

        float bv = -1.0f;
        int   bi = 0x7fffffff;
#pragma unroll
        for (int j = 0; j < 16; ++j) {
            const float dx = px[j] - cx, dy = py[j] - cy, dz = pz[j] - cz;
            const float dd = dx * dx + dy * dy + dz * dz;
            pd[j] = fminf(pd[j], dd);
            if (pd[j] > bv) { bv = pd[j]; bi = tid + j * 256; }
        }
        // wave32 argmax reduce, first-index tie-break
#pragma unroll
        for (int m = 16; m >= 1; m >>= 1) {
            const float ov = __shfl_xor(bv, m);
            const int   oi = __shfl_xor(bi, m);
            if (ov > bv || (ov == bv && oi < bi)) { bv = ov; bi = oi; }
        }
        if (lane == 0) { rv[wv] = bv; ri[wv] = bi; }
        __syncthreads();
        if (tid == 0) {
            float mv = rv[0]; int mi = ri[0];
            for (int t = 1; t < 8; ++t)
                if (rv[t] > mv || (rv[t] == mv && ri[t] < mi)) { mv = rv[t]; mi = ri[t]; }
            sfar = mi;
        }
        // no extra barrier needed: only tid0 reads sfar; rv/ri rewritten after
        // the NEXT iteration's first barrier.
    }
}

// =====================================================================
// 2) prep: pad W0 (64x6 -> 64x8 with zeros), zero the 512-float stats block
// =====================================================================
__global__ __launch_bounds__(256)
void prep_kernel(const float* __restrict__ w0, float* __restrict__ w0p,
                 float* __restrict__ stats)
{
    for (int i = threadIdx.x; i < 512; i += 256) {
        const int o = i >> 3, c = i & 7;
        w0p[i]   = (c < 6) ? w0[o * 6 + c] : 0.0f;
        stats[i] = 0.0f;
    }
}

// =====================================================================
// 3) Ball query + grouping. One wave32 per query: ordered index compaction
//    via ballot + prefix popcount (matches "first NSAMP ascending indices
//    within radius, pad with first"). Writes X0 as [8][M] channel-major,
//    channels 0-2 = xyz - q, 3-5 = points, 6-7 = zero pad.
// =====================================================================
__global__ __launch_bounds__(256)
void group_kernel(const float* __restrict__ xyz, const float* __restrict__ points,
                  const float* __restrict__ new_xyz, float* __restrict__ X0)
{
    __shared__ int slots[8][NSAMP];
    const int tid  = threadIdx.x;
    const int lane = tid & 31;
    const int wv   = tid >> 5;
    const int g    = blockIdx.x * 8 + wv;      // query id 0..16383
    const int b    = g >> 10;
    const float* xb = xyz    + (size_t)b * NPTS * 3;
    const float* pb = points + (size_t)b * NPTS * 3;
    const float* q  = new_xyz + (size_t)g * 3;
    const float qx = q[0], qy = q[1], qz = q[2];

    int cnt = 0;
    for (int n0 = 0; n0 < NPTS && cnt < NSAMP; n0 += 32) {
        const int n = n0 + lane;
        const float dx = xb[n * 3 + 0] - qx;
        const float dy = xb[n * 3 + 1] - qy;
        const float dz = xb[n * 3 + 2] - qz;
        const float d2 = dx * dx + dy * dy + dz * dz;
        const bool  pr = (d2 <= RAD2);
        const unsigned mask = (unsigned)__ballot(pr);
        const int pos = cnt + __popc(mask & ((1u << lane) - 1u));
        if (pr && pos < NSAMP) slots[wv][pos] = n;
        cnt += __popc(mask);
    }
    __syncthreads();
    const int kk  = (cnt < NSAMP) ? cnt : NSAMP;   // >= 1 (query point itself)
    const int src = (lane < kk) ? slots[wv][lane] : slots[wv][0];

    const size_t m = (size_t)g * NSAMP + lane;
    const float gx = xb[src * 3 + 0] - qx;
    const float gy = xb[src * 3 + 1] - qy;
    const float gz = xb[src * 3 + 2] - qz;
    X0[(size_t)0 * MCOLS + m] = gx;
    X0[(size_t)1 * MCOLS + m] = gy;
    X0[(size_t)2 * MCOLS + m] = gz;
    X0[(size_t)3 * MCOLS + m] = pb[src * 3 + 0];
    X0[(size_t)4 * MCOLS + m] = pb[src * 3 + 1];
    X0[(size_t)5 * MCOLS + m] = pb[src * 3 + 2];
    X0[(size_t)6 * MCOLS + m] = 0.0f;
    X0[(size_t)7 * MCOLS + m] = 0.0f;
}

// =====================================================================
// WMMA GEMM: Y[O][M] = W[O][C] * norm(X[C][M]) + bias, fused per-channel
// sum/sumsq (for batch-norm stats). One wave per 16-col M-tile, covering
// all O-tiles so the B operand is reused across O/16 WMMAs per k-step.
// O is kept <= 64 (T <= 4) so per-wave state (acc+stats+bias = 128 VGPRs)
// never spills; 128-channel layers are run as two 64-channel launches.
// V_WMMA_F32_16X16X4_F32, wave32 layouts per ISA 7.12.2:
//   A (16x4): lanes0-15 hold K=0/1, lanes16-31 hold K=2/3
//   B (4x16): vgpr0 rows K=0|2, vgpr1 rows K=1|3 (lane halves)
//   C/D     : vgpr v -> rows v (lanes 0-15) and v+8 (lanes 16-31)
// =====================================================================
template <int C, int O, bool NORM, bool STATS, bool STORE>
__global__ __launch_bounds__(256)
void gemm_wmma_kernel(const float* __restrict__ W, const float* __restrict__ bias,
                      const float* __restrict__ X, const float* __restrict__ ns,
                      const float* __restrict__ nh, float* __restrict__ Y,
                      float* __restrict__ gsum, float* __restrict__ gsq)
{
    constexpr int T = O / 16;
    __shared__ float lsum[O];
    __shared__ float lsq[O];
    const int tid  = threadIdx.x;
    const int lane = tid & 31;
    const int wv   = tid >> 5;
    if (STATS) {
        for (int i = tid; i < O; i += 256) { lsum[i] = 0.0f; lsq[i] = 0.0f; }
        __syncthreads();
    }
    const int lh   = lane & 15;
    const int koff = (lane < 16) ? 0 : 2;
    const int rofs = (lane < 16) ? 0 : 8;

    float bo[T][8];
#pragma unroll
    for (int t = 0; t < T; ++t)
#pragma unroll
        for (int v = 0; v < 8; ++v) bo[t][v] = bias[t * 16 + v + rofs];

    float ss[T][8], sq[T][8];
#pragma unroll
    for (int t = 0; t < T; ++t)
#pragma unroll
        for (int v = 0; v < 8; ++v) { ss[t][v] = 0.0f; sq[t][v] = 0.0f; }

    const int nwaves = gridDim.x * 8;
    const int MT     = MCOLS / 16;
    for (int mt = blockIdx.x * 8 + wv; mt < MT; mt += nwaves) {
        const int mcol = mt * 16 + lh;
        v8f acc[T];
        const v8f zero = {0.f, 0.f, 0.f, 0.f, 0.f, 0.f, 0.f, 0.f};
#pragma unroll
        for (int t = 0; t < T; ++t) acc[t] = zero;

#pragma unroll
        for (int kc = 0; kc < C; kc += 4) {
            const int ch0 = kc + koff;
            float x0 = X[(size_t)ch0 * MCOLS + mcol];
            float x1 = X[(size_t)(ch0 + 1) * MCOLS + mcol];
            if (NORM) {
                x0 = fmaxf(ns[ch0] * x0 + nh[ch0], 0.0f);
                x1 = fmaxf(ns[ch0 + 1] * x1 + nh[ch0 + 1], 0.0f);
            }
            const v2f bb = {x0, x1};
#pragma unroll
            for (int t = 0; t < T; ++t) {
                const int orow = t * 16 + lh;
                const v2f aa = {W[orow * C + ch0], W[orow * C + ch0 + 1]};
                acc[t] = __builtin_amdgcn_wmma_f32_16x16x4_f32(
                    false, aa, false, bb, (short)0, acc[t], false, false);
            }
        }
#pragma unroll
        for (int t = 0; t < T; ++t)
#pragma unroll
            for (int v = 0; v < 8; ++v) {
                const float y = acc[t][v] + bo[t][v];
                if (STATS) { ss[t][v] += y; sq[t][v] += y * y; }
                if (STORE) Y[(size_t)(t * 16 + v + rofs) * MCOLS + mcol] = y;
            }
    }

    if (STATS) {
#pragma unroll
        for (int t = 0; t < T; ++t)
#pragma unroll
            for (int v = 0; v < 8; ++v) {
                float s = ss[t][v], q = sq[t][v];
#pragma unroll
                for (int m = 1; m < 16; m <<= 1) {
                    s += __shfl_xor(s, m);
                    q += __shfl_xor(q, m);
                }
                if (lh == 0) {
                    atomicAdd(&lsum[t * 16 + v + rofs], s);
                    atomicAdd(&lsq[t * 16 + v + rofs], q);
                }
            }
        __syncthreads();
        for (int i = tid; i < O; i += 256) {
            atomicAdd(&gsum[i], lsum[i]);
            atomicAdd(&gsq[i], lsq[i]);
        }
    }
}

// =====================================================================
// finalize BN: ns = gamma * rsqrt(var+eps), nh = beta - mean*ns
// =====================================================================
__global__ __launch_bounds__(128)
void finalize_kernel(const float* __restrict__ sum, const float* __restrict__ sq,
                     const float* __restrict__ gamma, const float* __restrict__ beta,
                     float* __restrict__ ns, float* __restrict__ nh,
                     int O, float invM)
{
    const int o = threadIdx.x;
    if (o < O) {
        const float mean = sum[o] * invM;
        const float var  = sq[o] * invM - mean * mean;
        const float r    = rsqrtf(var + EPSV);
        const float a    = gamma[o] * r;
        ns[o] = a;
        nh[o] = beta[o] - mean * a;
    }
}

// =====================================================================
// Final layer recompute + BN + ReLU + max over NSAMP + transposed store.
// One wave per group (b,s) and per 64-channel half (obase in {0,64}):
// 2 M-tiles x 4 O-tiles of WMMA, max folded in registers, half-wave
// shuffle max reduce, lanes 0/16 store the 64 channels.
// W, bias, ns2, nh2 pointers are pre-offset by the caller; obase only
// selects the output sub-range.
// =====================================================================
__global__ __launch_bounds__(256)
void gemm_max_kernel(const float* __restrict__ W, const float* __restrict__ bias,
                     const float* __restrict__ X, const float* __restrict__ ns1,
                     const float* __restrict__ nh1, const float* __restrict__ ns2,
                     const float* __restrict__ nh2, float* __restrict__ outp,
                     int obase)
{
    constexpr int C = 64, T = 4;
    const int tid  = threadIdx.x;
    const int lane = tid & 31;
    const int wv   = tid >> 5;
    const int g    = blockIdx.x * 8 + wv;       // 0..16383
    const int lh   = lane & 15;
    const int koff = (lane < 16) ? 0 : 2;
    const int rofs = (lane < 16) ? 0 : 8;

    float mx[T][8];
#pragma unroll
    for (int t = 0; t < T; ++t)
#pragma unroll
        for (int v = 0; v < 8; ++v) mx[t][v] = -3.4e38f;

#pragma unroll
    for (int half = 0; half < 2; ++half) {
        const int mcol = g * 32 + half * 16 + lh;
        v8f acc[T];
        const v8f zero = {0.f, 0.f, 0.f, 0.f, 0.f, 0.f, 0.f, 0.f};
#pragma unroll
        for (int t = 0; t < T; ++t) acc[t] = zero;
#pragma unroll
        for (int kc = 0; kc < C; kc += 4) {
            const int ch0 = kc + koff;
            float x0 = X[(size_t)ch0 * MCOLS + mcol];
            float x1 = X[(size_t)(ch0 + 1) * MCOLS + mcol];
            x0 = fmaxf(ns1[ch0] * x0 + nh1[ch0], 0.0f);
            x1 = fmaxf(ns1[ch0 + 1] * x1 + nh1[ch0 + 1], 0.0f);
            const v2f bb = {x0, x1};
#pragma unroll
            for (int t = 0; t < T; ++t) {
                const int orow = t * 16 + lh;
                const v2f aa = {W[orow * C + ch0], W[orow * C + ch0 + 1]};
                acc[t] = __builtin_amdgcn_wmma_f32_16x16x4_f32(
                    false, aa, false, bb, (short)0, acc[t], false, false);
            }
        }
#pragma unroll
        for (int t = 0; t < T; ++t)
#pragma unroll
            for (int v = 0; v < 8; ++v) {
                const int o   = t * 16 + v + rofs;
                const float y = acc[t][v] + bias[o];
                const float r = fmaxf(ns2[o] * y + nh2[o], 0.0f);
                mx[t][v] = fmaxf(mx[t][v], r);
            }
    }
#pragma unroll
    for (int t = 0; t < T; ++t)
#pragma unroll
        for (int v = 0; v < 8; ++v) {
#pragma unroll
            for (int m = 1; m < 16; m <<= 1)
                mx[t][v] = fmaxf(mx[t][v], __shfl_xor(mx[t][v], m));
        }
    if (lh == 0) {
        float* o = outp + (size_t)g * 128 + obase;
#pragma unroll
        for (int t = 0; t < T; ++t)
#pragma unroll
            for (int v = 0; v < 8; ++v) o[t * 16 + v + rofs] = mx[t][v];
    }
}

// =====================================================================
extern "C" void kernel_launch(void* const* d_in, const int* in_sizes, int n_in,
                              void* d_out, int out_size, void* d_ws, size_t ws_size,
                              hipStream_t stream)
{
    const float* xyz    = (const float*)d_in[0];
    const float* points = (const float*)d_in[1];
    const float* w0  = (const float*)d_in[2];
    const float* b0  = (const float*)d_in[3];
    const float* g0  = (const float*)d_in[4];
    const float* be0 = (const float*)d_in[5];
    const float* w1  = (const float*)d_in[6];
    const float* b1  = (const float*)d_in[7];
    const float* g1  = (const float*)d_in[8];
    const float* be1 = (const float*)d_in[9];
    const float* w2  = (const float*)d_in[10];
    const float* b2  = (const float*)d_in[11];
    const float* g2  = (const float*)d_in[12];
    const float* be2 = (const float*)d_in[13];

    float* out = (float*)d_out;
    float* ws  = (float*)d_ws;

    float* X0   = ws + OFF_X0;
    float* Y0   = ws + OFF_Y0;
    float* Y1   = ws + OFF_Y1;
    float* W0P  = ws + OFF_W0P;
    float* SUM0 = ws + OFF_SUM0; float* SQ0 = ws + OFF_SQ0;
    float* SUM1 = ws + OFF_SUM1; float* SQ1 = ws + OFF_SQ1;
    float* SUM2 = ws + OFF_SUM2; float* SQ2 = ws + OFF_SQ2;
    float* NS0 = ws + OFF_NS0; float* NH0 = ws + OFF_NH0;
    float* NS1 = ws + OFF_NS1; float* NH1 = ws + OFF_NH1;
    float* NS2 = ws + OFF_NS2; float* NH2 = ws + OFF_NH2;

    float* new_xyz = out;                                   // 16*1024*3
    float* new_pts = out + (size_t)NBATCH * NPOINTS * 3;    // 16*1024*128

    const float invM = 1.0f / (float)MCOLS;

    fps_kernel<<<NBATCH, 256, 0, stream>>>(xyz, new_xyz);
    prep_kernel<<<1, 256, 0, stream>>>(w0, W0P, SUM0);
    group_kernel<<<2048, 256, 0, stream>>>(xyz, points, new_xyz, X0);

    // layer 0: [64x8pad] x [8xM]
    gemm_wmma_kernel<8, 64, false, true, true><<<512, 256, 0, stream>>>(
        W0P, b0, X0, nullptr, nullptr, Y0, SUM0, SQ0);
    finalize_kernel<<<1, 64, 0, stream>>>(SUM0, SQ0, g0, be0, NS0, NH0, 64, invM);

    // layer 1: [64x64] x norm([64xM])
    gemm_wmma_kernel<64, 64, true, true, true><<<512, 256, 0, stream>>>(
        w1, b1, Y0, NS0, NH0, Y1, SUM1, SQ1);
    finalize_kernel<<<1, 64, 0, stream>>>(SUM1, SQ1, g1, be1, NS1, NH1, 64, invM);

    // layer 2 stats: [128x64] x norm([64xM]) as two 64-channel halves
    // (keeps per-wave register state at T=4 -> no scratch spills)
    gemm_wmma_kernel<64, 64, true, true, false><<<512, 256, 0, stream>>>(
        w2, b2, Y1, NS1, NH1, nullptr, SUM2, SQ2);
    gemm_wmma_kernel<64, 64, true, true, false><<<512, 256, 0, stream>>>(
        w2 + 64 * 64, b2 + 64, Y1, NS1, NH1, nullptr, SUM2 + 64, SQ2 + 64);
    finalize_kernel<<<1, 128, 0, stream>>>(SUM2, SQ2, g2, be2, NS2, NH2, 128, invM);

    // layer 2 recompute + BN + ReLU + max over NSAMP, two 64-channel halves
    gemm_max_kernel<<<2048, 256, 0, stream>>>(
        w2, b2, Y1, NS1, NH1, NS2, NH2, new_pts, 0);
    gemm_max_kernel<<<2048, 256, 0, stream>>>(
        w2 + 64 * 64, b2 + 64, Y1, NS1, NH1, NS2 + 64, NH2 + 64, new_pts, 64);
}